// GraphNet_57432302682564
// MI455X (gfx1250) — compile-verified
//
#include <hip/hip_runtime.h>
#include <hip/hip_bf16.h>
#include <math.h>

typedef __attribute__((ext_vector_type(2))) float v2f;
typedef __attribute__((ext_vector_type(8))) float v8f;

// ---------------------------------------------------------------------------
// Utility kernels
// ---------------------------------------------------------------------------
__global__ void zero_kernel(float* __restrict__ p, long n) {
    long i = (long)blockIdx.x * blockDim.x + threadIdx.x;
    if (i < n) p[i] = 0.0f;
}

// h0[i, 0:3] = x[i], h0[i, 3:8] = tanh(i * w_pos + b_pos)
__global__ void build_h0_kernel(const float* __restrict__ x,
                                const float* __restrict__ w_pos,
                                const float* __restrict__ b_pos,
                                float* __restrict__ h0, int n) {
    int i = blockIdx.x * blockDim.x + threadIdx.x;
    if (i >= n) return;
    float fi = (float)i;
    float* r = h0 + (size_t)i * 8;
    r[0] = x[i * 3 + 0];
    r[1] = x[i * 3 + 1];
    r[2] = x[i * 3 + 2];
#pragma unroll
    for (int j = 0; j < 5; ++j) r[3 + j] = tanhf(fi * w_pos[j] + b_pos[j]);
}

// in-degree count (as float, for later reciprocal)
__global__ void count_kernel(const int* __restrict__ dst, float* __restrict__ cnt, int ne) {
    int e = blockIdx.x * blockDim.x + threadIdx.x;
    if (e >= ne) return;
    atomicAdd(&cnt[dst[e]], 1.0f);
}

// cnt[i] -> 1 / max(cnt[i], 1)   (in place)
__global__ void recip_kernel(float* __restrict__ cnt, int n) {
    int i = blockIdx.x * blockDim.x + threadIdx.x;
    if (i >= n) return;
    float c = cnt[i];
    cnt[i] = 1.0f / fmaxf(c, 1.0f);
}

// ---------------------------------------------------------------------------
// Edge scatter-add: agg[dst] += h[src]   (optionally only for dst < LIMIT)
// ---------------------------------------------------------------------------
template <int D, int LIMIT>
__global__ void agg_edges_kernel(const int* __restrict__ src, const int* __restrict__ dst,
                                 const float* __restrict__ h, float* __restrict__ agg, int ne) {
    int e = blockIdx.x * blockDim.x + threadIdx.x;
    if (e >= ne) return;
    int d = dst[e];
    if (LIMIT > 0 && d >= LIMIT) return;
    int s = src[e];
    const float4* hp = reinterpret_cast<const float4*>(h + (size_t)s * D);
    float* ap = agg + (size_t)d * D;
#pragma unroll
    for (int q = 0; q < D / 4; ++q) {
        float4 v = hp[q];  // global_load_b128
        atomicAdd(ap + q * 4 + 0, v.x);
        atomicAdd(ap + q * 4 + 1, v.y);
        atomicAdd(ap + q * 4 + 2, v.z);
        atomicAdd(ap + q * 4 + 3, v.w);
    }
}

// ---------------------------------------------------------------------------
// Dense SAGE transform via V_WMMA_F32_16X16X4_F32:
//   out = (agg * inv) @ Wl + h @ Wr + bias
// One wave computes a 16-row tile. K = DIN folded in chunks of 4.
// A layout (16x4 f32): lane l holds row M = l&15, K = 4c + 2*(l>=16) + {0,1}
// B layout (4x16 f32): lane l holds col N = l&15, same K split
// C/D layout (16x16 f32): VGPR r holds (M = r + 8*(l>=16), N = l&15)
// ---------------------------------------------------------------------------
template <int DIN, int DOUT, bool FINAL>
__global__ void sage_dense_kernel(const float* __restrict__ agg, const float* __restrict__ inv,
                                  const float* __restrict__ h,
                                  const float* __restrict__ Wl, const float* __restrict__ Wr,
                                  const float* __restrict__ bias, float* __restrict__ out,
                                  int ntiles) {
    int wave = (int)((blockIdx.x * blockDim.x + threadIdx.x) >> 5);
    if (wave >= ntiles) return;  // wave-uniform guard: EXEC all-1s below
    int lane = threadIdx.x & 31;
    int half = lane >> 4;  // 0: lanes 0-15, 1: lanes 16-31
    int nn = lane & 15;    // A-row M and B-col N for this lane

    int arow = wave * 16 + nn;
    float rinv = inv[arow];
    const float* aggrow = agg + (size_t)arow * DIN;
    const float* hrow = h + (size_t)arow * DIN;

    v8f acc = {};
    constexpr int CH = DIN / 4;

    // mean @ Wl
#pragma unroll
    for (int c = 0; c < CH; ++c) {
        int k0 = c * 4 + half * 2;
        v2f a, b;
        a.x = aggrow[k0] * rinv;
        a.y = aggrow[k0 + 1] * rinv;
        if (DOUT == 16) {
            b.x = Wl[k0 * DOUT + nn];
            b.y = Wl[(k0 + 1) * DOUT + nn];
        } else {
            b.x = (nn < DOUT) ? Wl[k0 * DOUT + nn] : 0.0f;
            b.y = (nn < DOUT) ? Wl[(k0 + 1) * DOUT + nn] : 0.0f;
        }
        acc = __builtin_amdgcn_wmma_f32_16x16x4_f32(false, a, false, b, (short)0, acc, false, false);
    }
    // h @ Wr
#pragma unroll
    for (int c = 0; c < CH; ++c) {
        int k0 = c * 4 + half * 2;
        v2f a, b;
        a.x = hrow[k0];
        a.y = hrow[k0 + 1];
        if (DOUT == 16) {
            b.x = Wr[k0 * DOUT + nn];
            b.y = Wr[(k0 + 1) * DOUT + nn];
        } else {
            b.x = (nn < DOUT) ? Wr[k0 * DOUT + nn] : 0.0f;
            b.y = (nn < DOUT) ? Wr[(k0 + 1) * DOUT + nn] : 0.0f;
        }
        acc = __builtin_amdgcn_wmma_f32_16x16x4_f32(false, a, false, b, (short)0, acc, false, false);
    }

    float bv = (nn < DOUT) ? bias[nn] : 0.0f;
#pragma unroll
    for (int r = 0; r < 8; ++r) {
        int orow = wave * 16 + r + half * 8;
        float v = acc[r] + bv;
        if (FINAL) {
            if (orow < 68 && nn < DOUT) out[orow * DOUT + nn] = v;
        } else {
            out[(size_t)orow * DOUT + nn] = v;
        }
    }
}

// ---------------------------------------------------------------------------
// Launch
// ---------------------------------------------------------------------------
extern "C" void kernel_launch(void* const* d_in, const int* in_sizes, int n_in,
                              void* d_out, int out_size, void* d_ws, size_t ws_size,
                              hipStream_t stream) {
    const float* x = (const float*)d_in[0];
    const int* ei = (const int*)d_in[1];
    const float* w_pos = (const float*)d_in[2];
    const float* b_pos = (const float*)d_in[3];
    const float* Wl1 = (const float*)d_in[4];
    const float* Wr1 = (const float*)d_in[5];
    const float* b1 = (const float*)d_in[6];
    const float* Wl2 = (const float*)d_in[7];
    const float* Wr2 = (const float*)d_in[8];
    const float* b2 = (const float*)d_in[9];
    const float* Wl3 = (const float*)d_in[10];
    const float* Wr3 = (const float*)d_in[11];
    const float* b3 = (const float*)d_in[12];
    float* out = (float*)d_out;

    const int N_ = in_sizes[0] / 3;   // 150000
    const int E_ = in_sizes[1] / 2;   // 4800000
    const int* src = ei;
    const int* dst = ei + E_;

    float* ws = (float*)d_ws;
    size_t Ns = (size_t)N_;
    float* cnt = ws;               // N  (becomes inv after recip)
    float* h0 = ws + Ns;           // 8N
    float* h1 = h0 + 8 * Ns;       // 16N
    float* h2 = h1 + 16 * Ns;      // 16N
    float* agg = h2 + 16 * Ns;     // 16N

    const int B = 256;
    auto blocks = [](long n) { return (unsigned)((n + 255) / 256); };

    // Stage 0: counts + positional embedding
    zero_kernel<<<blocks(Ns), B, 0, stream>>>(cnt, (long)Ns);
    zero_kernel<<<blocks(16 * Ns), B, 0, stream>>>(agg, (long)(16 * Ns));
    build_h0_kernel<<<blocks(Ns), B, 0, stream>>>(x, w_pos, b_pos, h0, N_);
    count_kernel<<<blocks(E_), B, 0, stream>>>(dst, cnt, E_);
    recip_kernel<<<blocks(Ns), B, 0, stream>>>(cnt, N_);

    const int ntiles = N_ / 16;  // 150000 / 16 = 9375 (exact)

    // Layer 1: d = 8 -> 16
    agg_edges_kernel<8, 0><<<blocks(E_), B, 0, stream>>>(src, dst, h0, agg, E_);
    sage_dense_kernel<8, 16, false><<<blocks((long)ntiles * 32), B, 0, stream>>>(
        agg, cnt, h0, Wl1, Wr1, b1, h1, ntiles);

    // Layer 2: d = 16 -> 16
    zero_kernel<<<blocks(16 * Ns), B, 0, stream>>>(agg, (long)(16 * Ns));
    agg_edges_kernel<16, 0><<<blocks(E_), B, 0, stream>>>(src, dst, h1, agg, E_);
    sage_dense_kernel<16, 16, false><<<blocks((long)ntiles * 32), B, 0, stream>>>(
        agg, cnt, h1, Wl2, Wr2, b2, h2, ntiles);

    // Layer 3: d = 16 -> 3, only rows < 68 are needed
    zero_kernel<<<blocks(16 * 80), B, 0, stream>>>(agg, (long)(16 * 80));
    agg_edges_kernel<16, 68><<<blocks(E_), B, 0, stream>>>(src, dst, h2, agg, E_);
    const int ftiles = (68 + 15) / 16;  // 5 tiles cover rows 0..79
    sage_dense_kernel<16, 3, true><<<blocks((long)ftiles * 32), B, 0, stream>>>(
        agg, cnt, h2, Wl3, Wr3, b3, out, ftiles);
}